// GNNAutoencoder_59184649339044
// MI455X (gfx1250) — compile-verified
//
#include <hip/hip_runtime.h>
#include <math.h>

typedef __attribute__((ext_vector_type(2))) float v2f;
typedef __attribute__((ext_vector_type(4))) float v4f;
typedef __attribute__((ext_vector_type(8))) float v8f;

// LDS weight tile stride: 68 floats -> bank = (4*row + col) % 64, so the
// 16 lanes of a B-fragment read (16 rows, same column pair) are conflict-free.
#define WSTRIDE 68

// ---------------------------------------------------------------------------
// fp32 WMMA 16x16x4 wrapper (CDNA5, wave32).
// A: 16x4 f32 (2 VGPRs)  lane l: row = l&15, holds K = 2*(l>>4) .. +1
// B: 4x16 f32 (2 VGPRs)  lane l: col = l&15, holds K = 2*(l>>4) .. +1
// C/D: 16x16 f32 (8 VGPRs) lane l, vgpr r: M = r + 8*(l>>4), N = l&15
// ---------------------------------------------------------------------------
__device__ __forceinline__ v8f wmma_f32_16x16x4(v2f a, v2f b, v8f c) {
  return __builtin_amdgcn_wmma_f32_16x16x4_f32(
      /*neg_a=*/false, a, /*neg_b=*/false, b,
      /*c_mod=*/(short)0, c, /*reuse_a=*/false, /*reuse_b=*/false);
}

#define ZERO8 {0.f,0.f,0.f,0.f,0.f,0.f,0.f,0.f}

// cooperative 64x64 weight load global -> padded LDS (float4 per thread x4)
__device__ __forceinline__ void load_w_to_lds(const float* __restrict__ W,
                                              float* __restrict__ Ws,
                                              int tid, int nthreads) {
  for (int i4 = tid; i4 < 1024; i4 += nthreads) {       // 1024 float4s
    v4f v = *(const v4f*)(W + (size_t)i4 * 4);
    int r = i4 >> 4;            // row (16 float4 per row)
    int c = (i4 & 15) * 4;      // col
    *(v4f*)(Ws + r * WSTRIDE + c) = v;                  // 16B aligned
  }
}

// ---------------------------------------------------------------------------
// init: h = 0, nf = 0, deg = 1 (self loop contributes 1 to every node degree)
// ---------------------------------------------------------------------------
__global__ void init_kernel(float* __restrict__ h, float* __restrict__ nf,
                            float* __restrict__ deg, int N) {
  size_t t = (size_t)blockIdx.x * blockDim.x + threadIdx.x;
  size_t n64 = (size_t)N * 64;
  if (t < n64) { h[t] = 0.f; nf[t] = 0.f; }
  if (t < (size_t)N) deg[t] = 1.0f;
}

// ---------------------------------------------------------------------------
// Generic out[M,64] = A[M,64] @ W[64,64]^T, one wave per 16-row strip.
// W staged in LDS once per block (8 waves share); B-fragments via ds_load_b64.
// K fully unrolled: 16 k-steps x 4 N-tiles = 64 v_wmma per strip.
// ---------------------------------------------------------------------------
template <bool NT>
__global__ void gemm_n64_nt(const float* __restrict__ A,
                            const float* __restrict__ W,
                            float* __restrict__ out, int M) {
  __shared__ float Ws[64 * WSTRIDE];
  load_w_to_lds(W, Ws, threadIdx.x, blockDim.x);
  __syncthreads();

  const int lane = threadIdx.x & 31;
  const int wid  = blockIdx.x * (blockDim.x >> 5) + (threadIdx.x >> 5);
  const int m0   = wid * 16;
  if (m0 >= M) return;                       // wave-uniform (after barrier)
  const int hi = lane >> 4;
  const int lo = lane & 15;

  int ar = m0 + lo; if (ar >= M) ar = M - 1; // clamp (EXEC stays all-ones)
  const float* arow = A + (size_t)ar * 64;

  v8f z = ZERO8;
  v8f acc[4] = {z, z, z, z};

#pragma unroll
  for (int k0 = 0; k0 < 64; k0 += 4) {
    v2f a = *(const v2f*)(arow + k0 + 2 * hi);
#pragma unroll
    for (int t = 0; t < 4; ++t) {
      v2f b = *(const v2f*)(Ws + (t * 16 + lo) * WSTRIDE + k0 + 2 * hi);
      acc[t] = wmma_f32_16x16x4(a, b, acc[t]);
    }
  }

#pragma unroll
  for (int t = 0; t < 4; ++t)
#pragma unroll
    for (int r = 0; r < 8; ++r) {
      int m = m0 + r + 8 * hi;
      if (m < M) {
        float* p = out + (size_t)m * 64 + t * 16 + lo;
        if (NT) __builtin_nontemporal_store(acc[t][r], p);
        else    *p = acc[t][r];
      }
    }
}

// ---------------------------------------------------------------------------
// degree count over edge targets (self loops already folded into init)
// ---------------------------------------------------------------------------
__global__ void deg_kernel(const int* __restrict__ col, float* __restrict__ deg,
                           int E) {
  int e = blockIdx.x * blockDim.x + threadIdx.x;
  if (e < E) atomicAdd(&deg[col[e]], 1.0f);
}

__global__ void dinv_kernel(const float* __restrict__ deg,
                            float* __restrict__ dinv, int N) {
  int n = blockIdx.x * blockDim.x + threadIdx.x;
  if (n < N) {
    float d = deg[n];
    dinv[n] = d > 0.f ? 1.0f / sqrtf(d) : 0.f;
  }
}

// ---------------------------------------------------------------------------
// GCN message scatter: h[col] += dinv[row]*dinv[col] * xw[row]
// 16 threads per edge, 4 channels each (float4 gather, 4 f32 atomics).
// ---------------------------------------------------------------------------
__global__ void msg_kernel(const float* __restrict__ xw,
                           const float* __restrict__ dinv,
                           const int* __restrict__ row,
                           const int* __restrict__ col,
                           float* __restrict__ h, int E) {
  size_t t = (size_t)blockIdx.x * blockDim.x + threadIdx.x;
  int e = (int)(t >> 4);
  if (e >= E) return;
  int j = (int)(t & 15) * 4;
  int r = row[e], c = col[e];
  float coef = dinv[r] * dinv[c];
  v4f v = *(const v4f*)(xw + (size_t)r * 64 + j);
  float* dst = h + (size_t)c * 64 + j;
  atomicAdd(dst + 0, coef * v[0]);
  atomicAdd(dst + 1, coef * v[1]);
  atomicAdd(dst + 2, coef * v[2]);
  atomicAdd(dst + 3, coef * v[3]);
}

// self-loop message + bias:  h[n] += dinv[n]^2 * xw[n] + b_mp
__global__ void selfloop_bias_kernel(float* __restrict__ h,
                                     const float* __restrict__ xw,
                                     const float* __restrict__ dinv,
                                     const float* __restrict__ b_mp, int N) {
  size_t t = (size_t)blockIdx.x * blockDim.x + threadIdx.x;
  if (t >= (size_t)N * 64) return;
  int n = (int)(t >> 6);
  int j = (int)(t & 63);
  float di = dinv[n];
  h[t] += di * di * xw[t] + b_mp[j];
}

// ---------------------------------------------------------------------------
// Wc = W_e @ W_n2e ; Wd = W_e2n @ Wc   (64x64 each, one block)
// ---------------------------------------------------------------------------
__global__ void combine_weights(const float* __restrict__ W_n2e,
                                const float* __restrict__ W_e,
                                const float* __restrict__ W_e2n,
                                float* __restrict__ Wc, float* __restrict__ Wd) {
  __shared__ float sWc[64 * 64];
  for (int idx = threadIdx.x; idx < 4096; idx += blockDim.x) {
    int o = idx >> 6, c = idx & 63;
    float s = 0.f;
    for (int k = 0; k < 64; ++k) s += W_e[o * 64 + k] * W_n2e[k * 64 + c];
    sWc[idx] = s;
    Wc[idx] = s;
  }
  __syncthreads();
  for (int idx = threadIdx.x; idx < 4096; idx += blockDim.x) {
    int o = idx >> 6, c = idx & 63;
    float s = 0.f;
    for (int k = 0; k < 64; ++k) s += W_e2n[o * 64 + k] * sWc[k * 64 + c];
    Wd[idx] = s;
  }
}

// ---------------------------------------------------------------------------
// Fused edge kernel: one wave = 16 edges.
//   gathered A = h[row[e0..e0+15]]  (one b64 global load per lane per k-step)
//   edge_features tile = A @ Wc^T   -> NT-stored to d_out edge section
//   ef tile            = A @ Wd^T   -> atomic scatter into nf[row]
// Wc/Wd staged in LDS per block; same A fragments feed both products:
// 128 v_wmma per 16 edges, B traffic on the ds pipe, A/atomics on vmem.
// ---------------------------------------------------------------------------
__global__ void edge_kernel(const float* __restrict__ h,
                            const int* __restrict__ row,
                            const float* __restrict__ Wc,
                            const float* __restrict__ Wd,
                            float* __restrict__ edge_out,
                            float* __restrict__ nf, int E) {
  __shared__ float Wcs[64 * WSTRIDE];
  __shared__ float Wds[64 * WSTRIDE];
  load_w_to_lds(Wc, Wcs, threadIdx.x, blockDim.x);
  load_w_to_lds(Wd, Wds, threadIdx.x, blockDim.x);
  __syncthreads();

  const int lane = threadIdx.x & 31;
  const int wid  = blockIdx.x * (blockDim.x >> 5) + (threadIdx.x >> 5);
  const int e0   = wid * 16;
  if (e0 >= E) return;                        // wave-uniform (after barrier)
  const int hi = lane >> 4;
  const int lo = lane & 15;

  int ea = e0 + lo; if (ea >= E) ea = E - 1;  // clamp, keep EXEC all-ones
  const float* arow = h + (size_t)row[ea] * 64;

  v8f z = ZERO8;
  v8f accC[4] = {z, z, z, z};                 // edge_features
  v8f accD[4] = {z, z, z, z};                 // ef (to scatter)

#pragma unroll
  for (int k0 = 0; k0 < 64; k0 += 4) {
    v2f a = *(const v2f*)(arow + k0 + 2 * hi);
#pragma unroll
    for (int t = 0; t < 4; ++t) {
      v2f bc = *(const v2f*)(Wcs + (t * 16 + lo) * WSTRIDE + k0 + 2 * hi);
      accC[t] = wmma_f32_16x16x4(a, bc, accC[t]);
      v2f bd = *(const v2f*)(Wds + (t * 16 + lo) * WSTRIDE + k0 + 2 * hi);
      accD[t] = wmma_f32_16x16x4(a, bd, accD[t]);
    }
  }

  int rid[8];
#pragma unroll
  for (int r = 0; r < 8; ++r) {
    int e = e0 + r + 8 * hi;
    rid[r] = row[e < E ? e : E - 1];
  }

#pragma unroll
  for (int t = 0; t < 4; ++t)
#pragma unroll
    for (int r = 0; r < 8; ++r) {
      int e = e0 + r + 8 * hi;
      if (e < E) {
        // streaming write-once output: non-temporal
        __builtin_nontemporal_store(accC[t][r],
                                    &edge_out[(size_t)e * 64 + t * 16 + lo]);
        atomicAdd(&nf[(size_t)rid[r] * 64 + t * 16 + lo], accD[t][r]);
      }
    }
}

// ---------------------------------------------------------------------------
// host-side launcher
// ---------------------------------------------------------------------------
extern "C" void kernel_launch(void* const* d_in, const int* in_sizes, int n_in,
                              void* d_out, int out_size, void* d_ws,
                              size_t ws_size, hipStream_t stream) {
  const float* x       = (const float*)d_in[0];
  const int*   eidx    = (const int*)d_in[1];
  const float* W_mp    = (const float*)d_in[2];
  const float* b_mp    = (const float*)d_in[3];
  const float* W_n2e   = (const float*)d_in[4];
  const float* W_e     = (const float*)d_in[5];
  const float* W_e2n   = (const float*)d_in[6];
  const float* W_recon = (const float*)d_in[7];

  const int N = in_sizes[0] / 64;
  const int E = in_sizes[1] / 2;
  const int* row = eidx;       // edge_index[0]
  const int* col = eidx + E;   // edge_index[1]

  const size_t n64 = (size_t)N * 64;

  // workspace layout (floats)
  float* ws   = (float*)d_ws;
  float* xw   = ws;              // N*64
  float* h    = xw + n64;        // N*64
  float* nf   = h + n64;         // N*64
  float* deg  = nf + n64;        // N
  float* dinv = deg + N;         // N
  float* Wc   = dinv + N;        // 64*64
  float* Wd   = Wc + 4096;       // 64*64

  float* x_recon  = (float*)d_out;          // [N,64]
  float* edge_out = (float*)d_out + n64;    // [E,64]

  const int ntiles = (N + 15) / 16;
  const int etiles = (E + 15) / 16;

  // 1) zero accumulators, deg=1 (self loops)
  init_kernel<<<(unsigned)((n64 + 255) / 256), 256, 0, stream>>>(h, nf, deg, N);
  // 2) xw = x @ W_mp^T  (xw is re-read by the scatter: keep cached)
  gemm_n64_nt<false><<<(ntiles + 7) / 8, 256, 0, stream>>>(x, W_mp, xw, N);
  // 3) degree of targets, then dinv = deg^-1/2
  deg_kernel<<<(E + 255) / 256, 256, 0, stream>>>(col, deg, E);
  dinv_kernel<<<(N + 255) / 256, 256, 0, stream>>>(deg, dinv, N);
  // 4) normalized message scatter + self-loop/bias fixup
  msg_kernel<<<(unsigned)(((size_t)E * 16 + 255) / 256), 256, 0, stream>>>(
      xw, dinv, row, col, h, E);
  selfloop_bias_kernel<<<(unsigned)((n64 + 255) / 256), 256, 0, stream>>>(
      h, xw, dinv, b_mp, N);
  // 5) fold the three 64x64 edge-path weights into Wc, Wd
  combine_weights<<<1, 256, 0, stream>>>(W_n2e, W_e, W_e2n, Wc, Wd);
  // 6) fused edge gather + dual GEMM + scatter
  edge_kernel<<<(etiles + 7) / 8, 256, 0, stream>>>(h, row, Wc, Wd, edge_out,
                                                    nf, E);
  // 7) x_recon = nf @ W_recon^T  (write-once output: non-temporal)
  gemm_n64_nt<true><<<(ntiles + 7) / 8, 256, 0, stream>>>(nf, W_recon,
                                                          x_recon, N);
}